// MBGCN_63187558859193
// MI455X (gfx1250) — compile-verified
//
#include <hip/hip_runtime.h>
#include <hip/hip_bf16.h>

typedef __attribute__((ext_vector_type(2))) float v2f;
typedef __attribute__((ext_vector_type(8))) float v8f;

#define EPSF 1e-8f

__device__ __forceinline__ void atomAddF(float* p, float v) {
#if defined(__AMDGCN__)
  unsafeAtomicAdd(p, v);   // hardware global_atomic_add_f32
#else
  atomicAdd(p, v);
#endif
}

// ---------------- utility kernels ----------------

__global__ void zero_f32(float* __restrict__ p, long long n) {
  long long t = (long long)blockIdx.x * blockDim.x + threadIdx.x;
  long long stride = (long long)gridDim.x * blockDim.x;
  for (; t < n; t += stride) p[t] = 0.0f;
}

__global__ void compute_ubw(const float* __restrict__ ubd, const float* __restrict__ mg,
                            float* __restrict__ ubw, int U, int R) {
  int u = blockIdx.x * blockDim.x + threadIdx.x;
  if (u >= U) return;
  float tw = 0.f;
  for (int r = 0; r < R; ++r) tw += ubd[u * R + r] * mg[r];
  float inv = 1.0f / (tw + EPSF);
  for (int r = 0; r < R; ++r) ubw[u * R + r] = ubd[u * R + r] * mg[r] * inv;
}

// Pair-pack a [KD, NN] row-major matrix for WMMA SRC1 fragments:
//   P[(k>>1)*NN*2 + n*2 + (k&1)] = B[k*NN + n]
// so {B[k][n], B[k+1][n]} (k even) is one aligned 8-byte pair -> one ds_load_b64.
__global__ void pack_pairs(const float* __restrict__ B, float* __restrict__ P, int KD, int NN) {
  int t = blockIdx.x * blockDim.x + threadIdx.x;
  if (t >= KD * NN) return;
  int k = t / NN, n = t % NN;
  P[((((k >> 1) * NN) + n) << 1) | (k & 1)] = B[t];
}

// sparse [n_out, n_in] @ dense [n_in, D] -> out [n_out, D], via gather + atomic scatter.
// D consecutive lanes handle one edge => coalesced load + coalesced atomic add.
template <int D>
__global__ void spmm_atomic(const int* __restrict__ rows, const int* __restrict__ cols,
                            const float* __restrict__ vals,
                            const float* __restrict__ dense, float* __restrict__ out,
                            int nnz) {
  constexpr int EPB = 256 / D;
  int e = blockIdx.x * EPB + (threadIdx.x / D);
  int d = threadIdx.x % D;
  if (e >= nnz) return;
  int r = rows[e];
  int c = cols[e];
  float v = vals[e];
  atomAddF(out + (size_t)r * D + d, v * dense[(size_t)c * D + d]);
}

// agg /= (deg+eps)  (in place, becomes ig_n);  tip[:, :64] = item_embedding
__global__ void ig_norm_tip(float* __restrict__ agg, const float* __restrict__ deg,
                            const float* __restrict__ item_emb, float* __restrict__ tip, int I) {
  int t = blockIdx.x * blockDim.x + threadIdx.x;
  if (t >= I * 64) return;
  int i = t >> 6, d = t & 63;
  agg[t] = agg[t] / (deg[i] + EPSF);
  tip[((size_t)i << 7) + d] = item_emb[t];
}

// tuinp /= (ubd[:,rel]+eps); user_feature += ubw[:,rel] * tuinp[:, :64]  (tun == lower half)
__global__ void tuinp_norm_uf(float* __restrict__ tuinp, const float* __restrict__ ubd,
                              const float* __restrict__ ubw, float* __restrict__ user_feat,
                              int U, int R, int rel) {
  int t = blockIdx.x * blockDim.x + threadIdx.x;
  if (t >= U * 128) return;
  int u = t >> 7, c = t & 127;
  float v = tuinp[t] / (ubd[u * R + rel] + EPSF);
  tuinp[t] = v;
  if (c < 64) user_feat[(u << 6) + c] += ubw[u * R + rel] * v;
}

__global__ void gather_rows(const float* __restrict__ src, int ldsrc,
                            const int* __restrict__ idx, float* __restrict__ dst,
                            int lddst, int coff, int ncols, long long nrows) {
  long long t = (long long)blockIdx.x * blockDim.x + threadIdx.x;
  if (t >= nrows * ncols) return;
  long long r = t / ncols;
  int c = (int)(t % ncols);
  dst[r * lddst + coff + c] = src[(size_t)idx[r] * ldsrc + c];
}

// ---------------- WMMA fp32 GEMM:  C[M,NN] = A[M,KD] @ B[KD,NN] ----------------
// Bp is the pair-packed panel (see pack_pairs).  The whole panel is staged into
// LDS with CDNA5 async copies (global_load_async_to_lds_b128, ASYNCcnt-tracked),
// then each of the 8 waves holds a 16-row A strip in registers (16x4 fragments)
// and sweeps NN/16 column tiles K-stepping by 4 with V_WMMA_F32_16X16X4_F32.
// Every B fragment is a single aligned ds_load_b64 -> even VGPR pair (no repack
// movs).  All M used here are multiples of 16, so EXEC is all-1s at every WMMA;
// ragged trailing waves exit via a wave-uniform branch.
template <int KD, int NN>
__launch_bounds__(256)
__global__ void gemm_wmma(const float* __restrict__ A, int lda,
                          const float* __restrict__ Bp,
                          float* __restrict__ C, int ldc, int M) {
  extern __shared__ float sB[];
  const int tid = threadIdx.x;
#if defined(__AMDGCN__)
  {
    // straight contiguous copy of the packed panel: 16B per lane per iteration
    unsigned lbase = (unsigned)(size_t)(&sB[0]);
    for (int idx = tid * 4; idx < KD * NN; idx += 1024) {
      unsigned laddr = lbase + (unsigned)idx * 4u;
      unsigned long long gaddr = (unsigned long long)(size_t)(Bp + idx);
      asm volatile("global_load_async_to_lds_b128 %0, %1, off"
                   :: "v"(laddr), "v"(gaddr) : "memory");
    }
    asm volatile("s_wait_asynccnt 0x0" ::: "memory");
  }
#else
  for (int idx = tid; idx < KD * NN; idx += 256) sB[idx] = Bp[idx];
#endif
  __syncthreads();

  const int wave = tid >> 5;
  const int lane = tid & 31;
  const int m0 = (blockIdx.x * 8 + wave) * 16;
  if (m0 >= M) return;                 // wave-uniform exit

  const int mrow = lane & 15;          // A fragment: row = lane%16
  const int half = lane >> 4;          // lanes 0-15 -> k 0..1 ; lanes 16-31 -> k 2..3
  const int koff = half << 1;
  const int mo   = half << 3;          // C/D fragment: M base 0 / 8

  v2f areg[KD / 4];
  const float* Arow = A + (size_t)(m0 + mrow) * lda + koff;
#pragma unroll
  for (int s = 0; s < KD / 4; ++s)
    areg[s] = *(const v2f*)(Arow + (s << 2));

  for (int j = 0; j < NN; j += 16) {
    v8f acc = {};
#pragma unroll
    for (int s = 0; s < KD / 4; ++s) {
      const int p = (s << 1) + half;                    // k-pair index
      v2f b = *(const v2f*)&sB[(p * NN + j + mrow) << 1];
      acc = __builtin_amdgcn_wmma_f32_16x16x4_f32(false, areg[s], false, b,
                                                  (short)0, acc, false, false);
    }
    float* Cp = C + (size_t)(m0 + mo) * ldc + j + mrow;
#pragma unroll
    for (int v = 0; v < 8; ++v) Cp[(size_t)v * ldc] = acc[v];
  }
}

// ---------------- scoring ----------------

// score2[w] += dot(tbp_g[b,:], tip[item[w],:]) over 128, one wave per (b,k)
__global__ void score2_accum(const float* __restrict__ tbp_g, const float* __restrict__ tip,
                             const int* __restrict__ item, float* __restrict__ score2,
                             int BK, int K) {
  int w = (blockIdx.x * blockDim.x + threadIdx.x) >> 5;
  int lane = threadIdx.x & 31;
  if (w >= BK) return;
  int b = w / K;
  int it = item[w];
  float s = 0.f;
  for (int c = lane; c < 128; c += 32)
    s += tbp_g[(size_t)b * 128 + c] * tip[(size_t)it * 128 + c];
  for (int off = 16; off; off >>= 1) s += __shfl_down(s, off, 32);
  if (lane == 0) score2[w] += s;
}

// scores[w] = dot(uf_g[b], itf_g[w]) + (LAMB/R)*score2[w];  L2 += 1e-4*(|uf|^2+|itf|^2)
__global__ void final_score(const float* __restrict__ uf_g, const float* __restrict__ itf_g,
                            const float* __restrict__ score2, float* __restrict__ out,
                            int BK, int K, float lamb_over_R) {
  int w = (blockIdx.x * blockDim.x + threadIdx.x) >> 5;
  int lane = threadIdx.x & 31;
  if (w >= BK) return;
  int b = w / K;
  float s1 = 0.f, l2 = 0.f;
  for (int c = lane; c < 128; c += 32) {
    float u = uf_g[(size_t)b * 128 + c];
    float t = itf_g[(size_t)w * 128 + c];
    s1 += u * t;
    l2 += u * u + t * t;
  }
  for (int off = 16; off; off >>= 1) {
    s1 += __shfl_down(s1, off, 32);
    l2 += __shfl_down(l2, off, 32);
  }
  if (lane == 0) {
    out[w] = s1 + lamb_over_R * score2[w];
    atomAddF(out + BK, 1e-4f * l2);
  }
}

// ---------------- host ----------------

extern "C" void kernel_launch(void* const* d_in, const int* in_sizes, int n_in,
                              void* d_out, int out_size, void* d_ws, size_t ws_size,
                              hipStream_t stream) {
  const int*   user     = (const int*)d_in[0];
  const int*   item     = (const int*)d_in[1];
  const int*   rel_rows = (const int*)d_in[2];
  const int*   rel_cols = (const int*)d_in[3];
  const float* rel_vals = (const float*)d_in[4];
  const int*   ig_rows  = (const int*)d_in[5];
  const int*   ig_cols  = (const int*)d_in[6];
  const float* ig_vals  = (const float*)d_in[7];
  const int*   t_rows   = (const int*)d_in[8];
  const int*   t_cols   = (const int*)d_in[9];
  const float* t_vals   = (const float*)d_in[10];
  const float* ubd      = (const float*)d_in[11];
  const float* ig_deg   = (const float*)d_in[12];
  const float* user_emb = (const float*)d_in[13];
  const float* item_emb = (const float*)d_in[14];
  const float* mg       = (const float*)d_in[15];
  const float* Wp       = (const float*)d_in[16];
  const float* Wb       = (const float*)d_in[17];
  const float* W        = (const float*)d_in[18];

  const int R  = in_sizes[15];
  const int D  = 64, D2 = 128;
  const int U  = in_sizes[13] / D;
  const int I  = in_sizes[14] / D;
  const int B  = in_sizes[0];
  const int K  = in_sizes[1] / B;
  const int NNZ_R  = in_sizes[2] / R;
  const int NNZ_IG = in_sizes[5] / R;
  const int NNZ_T  = in_sizes[8];
  float* out = (float*)d_out;

  // workspace carve-up (floats); every region is a multiple of 4 floats -> 16B aligned
  float* w = (float*)d_ws;
  float* agg       = w; w += (size_t)I * D;        // ig aggregate -> ig_n
  float* tip       = w; w += (size_t)I * D2;       // [item_emb | ig_n @ Wp]
  float* tuinp     = w; w += (size_t)U * D2;       // rel spmm over tip, normalized
  float* user_feat = w; w += (size_t)U * D;
  float* item_feat = w; w += (size_t)I * D;
  float* ubw       = w; w += (size_t)U * R;
  float* tu_g      = w; w += (size_t)B * D2;       // tuinp gathered at users
  float* tbp_g     = w; w += (size_t)B * D2;       // tu_g @ Wb
  float* uf_g      = w; w += (size_t)B * D2;       // [user_emb | user_feat@W] gathered
  float* Au        = w; w += (size_t)B * D;
  float* itf_g     = w; w += (size_t)B * K * D2;   // [item_emb | item_feat@W] gathered
  float* Ai        = w; w += (size_t)B * K * D;
  float* score2    = w; w += (size_t)B * K;
  float* pWp       = w; w += (size_t)R * D * D;    // pair-packed weights for WMMA SRC1
  float* pWb       = w; w += (size_t)R * D2 * D2;
  float* pW        = w; w += (size_t)D * D;

  auto zero = [&](float* p, long long n) {
    long long blk = (n + 255) / 256;
    if (blk > 16384) blk = 16384;
    zero_f32<<<dim3((unsigned)blk), dim3(256), 0, stream>>>(p, n);
  };

  // pre-pack all GEMM weight panels once (reused by every gemm launch)
  for (int i = 0; i < R; ++i) {
    pack_pairs<<<(D * D + 255) / 256, 256, 0, stream>>>(
        Wp + (size_t)i * D * D, pWp + (size_t)i * D * D, D, D);
    pack_pairs<<<(D2 * D2 + 255) / 256, 256, 0, stream>>>(
        Wb + (size_t)i * D2 * D2, pWb + (size_t)i * D2 * D2, D2, D2);
  }
  pack_pairs<<<(D * D + 255) / 256, 256, 0, stream>>>(W, pW, D, D);

  zero(user_feat, (long long)U * D);
  zero(score2, (long long)B * K);
  compute_ubw<<<(U + 255) / 256, 256, 0, stream>>>(ubd, mg, ubw, U, R);

  for (int i = 0; i < R; ++i) {
    // ig = spmm(ig) / (deg+eps);  tip = [item_emb | ig @ Wp[i]]
    zero(agg, (long long)I * D);
    spmm_atomic<64><<<(NNZ_IG + 3) / 4, 256, 0, stream>>>(
        ig_rows + (size_t)i * NNZ_IG, ig_cols + (size_t)i * NNZ_IG,
        ig_vals + (size_t)i * NNZ_IG, item_emb, agg, NNZ_IG);
    ig_norm_tip<<<(I * 64 + 255) / 256, 256, 0, stream>>>(
        agg, ig_deg + (size_t)i * I, item_emb, tip, I);
    gemm_wmma<64, 64><<<(I + 127) / 128, 256, 64 * 64 * 4, stream>>>(
        agg, 64, pWp + (size_t)i * D * D, tip + 64, 128, I);

    // tuinp = spmm(rel, tip) / deg;  user_feature += ubw * tuinp[:, :64]
    zero(tuinp, (long long)U * D2);
    spmm_atomic<128><<<(NNZ_R + 1) / 2, 256, 0, stream>>>(
        rel_rows + (size_t)i * NNZ_R, rel_cols + (size_t)i * NNZ_R,
        rel_vals + (size_t)i * NNZ_R, tip, tuinp, NNZ_R);
    tuinp_norm_uf<<<(U * 128 + 255) / 256, 256, 0, stream>>>(
        tuinp, ubd, ubw, user_feat, U, R, i);

    // score2 += <(tuinp[user] @ Wb[i]), tip[item]>  -- gather rows first (1024 vs 50000)
    gather_rows<<<(B * D2 + 255) / 256, 256, 0, stream>>>(
        tuinp, 128, user, tu_g, 128, 0, 128, B);
    gemm_wmma<128, 128><<<(B + 127) / 128, 256, 128 * 128 * 4, stream>>>(
        tu_g, 128, pWb + (size_t)i * D2 * D2, tbp_g, 128, B);
    score2_accum<<<(B * K * 32 + 255) / 256, 256, 0, stream>>>(
        tbp_g, tip, item, score2, B * K, K);
  }

  // item_feature = train_matrix.T @ user_embedding  (rows = t_cols)
  zero(item_feat, (long long)I * D);
  spmm_atomic<64><<<(NNZ_T + 3) / 4, 256, 0, stream>>>(
      t_cols, t_rows, t_vals, user_emb, item_feat, NNZ_T);

  // uf_g = [user_emb[user] | user_feat[user] @ W]
  gather_rows<<<(B * D + 255) / 256, 256, 0, stream>>>(user_emb, 64, user, uf_g, 128, 0, 64, B);
  gather_rows<<<(B * D + 255) / 256, 256, 0, stream>>>(user_feat, 64, user, Au, 64, 0, 64, B);
  gemm_wmma<64, 64><<<(B + 127) / 128, 256, 64 * 64 * 4, stream>>>(Au, 64, pW, uf_g + 64, 128, B);

  // itf_g = [item_emb[item] | item_feat[item] @ W]
  gather_rows<<<(B * K * D + 255) / 256, 256, 0, stream>>>(item_emb, 64, item, itf_g, 128, 0, 64, (long long)B * K);
  gather_rows<<<(B * K * D + 255) / 256, 256, 0, stream>>>(item_feat, 64, item, Ai, 64, 0, 64, (long long)B * K);
  gemm_wmma<64, 64><<<(B * K + 127) / 128, 256, 64 * 64 * 4, stream>>>(Ai, 64, pW, itf_g + 64, 128, B * K);

  // scores + L2
  zero(out + (size_t)B * K, 1);
  final_score<<<(B * K * 32 + 255) / 256, 256, 0, stream>>>(
      uf_g, itf_g, score2, out, B * K, K, 0.5f / (float)R);
}